// PNN_498216206662
// MI455X (gfx1250) — compile-verified
//
#include <hip/hip_runtime.h>

// ---------------------------------------------------------------------------
// Gaussian-kernel regression (Nadaraya-Watson), MI455X / gfx1250, wave32.
//
//   pred = (W @ train_y) / (rowsum(W) + eps),  W[b,n] = exp(-dist2/200)
//
// Per-row constant exp(-||x||^2/200) cancels in the normalization, so
//   w'[b,n] = exp(cross[b,n]/100 - tsq[n]/200)
// Both GEMMs run on V_WMMA_F32_16X16X4_F32 (full fp32 precision).
//
// v2: each wave owns TWO 16-row x-tiles, so every train_x / train_y
// fragment load feeds two WMMAs (halves L2 traffic, doubles wmma:vmem).
// Train dim re-sliced 125x25 to keep ~8000 waves in flight.
// ---------------------------------------------------------------------------

typedef float v2f __attribute__((ext_vector_type(2)));
typedef float v8f __attribute__((ext_vector_type(8)));

#define B_ROWS   2048
#define N_TRAIN  50000
#define DIMS     64
#define COLS     16
#define SLICES   125           // 50000/16 = 3125 chunks = 125 slices * 25
#define CHUNKS_PER_SLICE 25

__device__ __forceinline__ float bperm(float v, int srcLane) {
    return __builtin_bit_cast(
        float, __builtin_amdgcn_ds_bpermute(srcLane << 2,
                                            __builtin_bit_cast(int, v)));
}

// --------------------------- kernel 0: row norms ---------------------------
__global__ void pnn_tsq_kernel(const float* __restrict__ tx,
                               float* __restrict__ tsq, int n_rows) {
    int n = blockIdx.x * blockDim.x + threadIdx.x;
    if (n >= n_rows) return;
    const float4* row = (const float4*)(tx + (size_t)n * DIMS);
    float s = 0.f;
#pragma unroll
    for (int i = 0; i < DIMS / 4; ++i) {
        float4 v = row[i];
        s += v.x * v.x + v.y * v.y + v.z * v.z + v.w * v.w;
    }
    tsq[n] = s;
}

// --------------------------- kernel 1: main --------------------------------
// 1 wave (32 threads) per block. blockIdx.x = 32-row x tile pair,
// blockIdx.y = train slice.
__global__ __launch_bounds__(32) void pnn_main_kernel(
    const float* __restrict__ x, const float* __restrict__ tx,
    const float* __restrict__ ty, const float* __restrict__ tsq,
    float* __restrict__ pred_acc, float* __restrict__ wsum_acc) {

    const int lane = threadIdx.x;      // 0..31
    const int lh   = lane & 15;
    const int h    = lane >> 4;        // half-wave select
    const int row0 = blockIdx.x * 32;  // two 16-row tiles: row0, row0+16
    const int chunk0 = blockIdx.y * CHUNKS_PER_SLICE;

    // Preload both x tiles as B-fragments of cross^T = T(16n x 64d) * X^T.
    // B frag (4x16, 32-bit): v0/v1 = cols {4c+2h, 4c+2h+1} of x row (base+lh).
    v2f xb0[16], xb1[16];
    {
        const float* xr0 = x + (size_t)(row0 + lh) * DIMS + 2 * h;
        const float* xr1 = xr0 + (size_t)16 * DIMS;
#pragma unroll
        for (int c = 0; c < 16; ++c) {
            xb0[c] = *(const v2f*)(xr0 + 4 * c);
            xb1[c] = *(const v2f*)(xr1 + 4 * c);
        }
    }

    v8f pred0 = {0.f, 0.f, 0.f, 0.f, 0.f, 0.f, 0.f, 0.f};
    v8f pred1 = pred0;
    float ws0 = 0.f, ws1 = 0.f;

    for (int ch = 0; ch < CHUNKS_PER_SLICE; ++ch) {
        const int n0 = (chunk0 + ch) * 16;
        const float* tr = tx + (size_t)(n0 + lh) * DIMS + 2 * h;

        if (ch + 1 < CHUNKS_PER_SLICE) {  // gfx1250 global_prefetch_b8
            __builtin_prefetch(tx + (size_t)(n0 + 16 + lh) * DIMS, 0, 1);
        }

        // ---- load this chunk's A-fragments once (shared by both tiles) ----
        // A frag (16x4, 32-bit): row (n0+lh), cols {4c+2h, 4c+2h+1}.
        v2f af[16];
#pragma unroll
        for (int c = 0; c < 16; ++c)
            af[c] = *(const v2f*)(tr + 4 * c);

        // ---- cross^T: D'[n][m] = sum_d T[n][d] * x[m][d], both tiles ------
        // Alternating acc0/acc1 on the same A spaces the accumulate chain.
        v8f acc0 = {0.f, 0.f, 0.f, 0.f, 0.f, 0.f, 0.f, 0.f};
        v8f acc1 = acc0;
#pragma unroll
        for (int c = 0; c < 16; ++c) {
            acc0 = __builtin_amdgcn_wmma_f32_16x16x4_f32(
                false, af[c], false, xb0[c], (short)0, acc0, false, false);
            acc1 = __builtin_amdgcn_wmma_f32_16x16x4_f32(
                false, af[c], false, xb1[c], (short)0, acc1, false, false);
        }

        // ---- weights: w'[n][m] = exp(cross/100 - tsq[n]/200) ---------------
        // D layout: VGPR i, lane l -> (n = i + 8h, m = lh). tsq uniform per
        // half-wave per VGPR: load tsq[n0 + 8h .. +7] as two float4s.
        const float4* tq = (const float4*)(tsq + n0 + 8 * h);
        float4 tq0 = tq[0], tq1 = tq[1];
        float tqv[8] = {tq0.x, tq0.y, tq0.z, tq0.w,
                        tq1.x, tq1.y, tq1.z, tq1.w};
        v8f wt0, wt1;
        float p0 = 0.f, p1 = 0.f;
#pragma unroll
        for (int i = 0; i < 8; ++i) {
            float b = tqv[i] * 0.005f;
            float w0 = __expf(acc0[i] * 0.01f - b);
            float w1 = __expf(acc1[i] * 0.01f - b);
            wt0[i] = w0; p0 += w0;
            wt1[i] = w1; p1 += w1;
        }
        ws0 += p0;
        ws1 += p1;

        // ---- second GEMM: pred(16m x 16c) += W(16m x 16n) * Y(16n x 16c) ---
        // A2 frag needs w'[m][4kk+2h+{0,1}]; w' lives transposed in wt's
        // D-layout, so gather with ds_bpermute (uniform src VGPR per half).
        // Y fragments are loaded once and feed both tiles' WMMAs.
        const float* yb = ty + (size_t)n0 * COLS + lh;
#pragma unroll
        for (int kk = 0; kk < 4; ++kk) {
            const int src = lh + ((kk >= 2) ? 16 : 0);  // + 16*(n/8)
            const int vlo = (4 * kk) & 7;               // 0,4,0,4
            v2f b2;
            b2.x = yb[(4 * kk + 2 * h + 0) * COLS];     // Y[n][c=lh]
            b2.y = yb[(4 * kk + 2 * h + 1) * COLS];

            v2f a2;
            {
                float ax_lo = bperm(wt0[vlo + 0], src); // n = 4kk   (h==0)
                float ay_lo = bperm(wt0[vlo + 1], src); // n = 4kk+1
                float ax_hi = bperm(wt0[vlo + 2], src); // n = 4kk+2 (h==1)
                float ay_hi = bperm(wt0[vlo + 3], src); // n = 4kk+3
                a2.x = h ? ax_hi : ax_lo;
                a2.y = h ? ay_hi : ay_lo;
            }
            pred0 = __builtin_amdgcn_wmma_f32_16x16x4_f32(
                false, a2, false, b2, (short)0, pred0, false, false);

            {
                float ax_lo = bperm(wt1[vlo + 0], src);
                float ay_lo = bperm(wt1[vlo + 1], src);
                float ax_hi = bperm(wt1[vlo + 2], src);
                float ay_hi = bperm(wt1[vlo + 3], src);
                a2.x = h ? ax_hi : ax_lo;
                a2.y = h ? ay_hi : ay_lo;
            }
            pred1 = __builtin_amdgcn_wmma_f32_16x16x4_f32(
                false, a2, false, b2, (short)0, pred1, false, false);
        }
    }

    // Fold weight-sums across half-waves: wsum[m] replicated at lanes m, m+16.
    ws0 += bperm(ws0, lane ^ 16);
    ws1 += bperm(ws1, lane ^ 16);

    // Accumulate partial results across train slices.
#pragma unroll
    for (int i = 0; i < 8; ++i) {
        int m = i + 8 * h;  // pred D-layout: (m = i + 8h, c = lh)
        atomicAdd(&pred_acc[(size_t)(row0 + m) * COLS + lh], pred0[i]);
        atomicAdd(&pred_acc[(size_t)(row0 + 16 + m) * COLS + lh], pred1[i]);
    }
    if (h == 0) {
        atomicAdd(&wsum_acc[row0 + lh], ws0);
        atomicAdd(&wsum_acc[row0 + 16 + lh], ws1);
    }
}

// --------------------------- kernel 2: finalize ----------------------------
__global__ void pnn_finalize_kernel(const float* __restrict__ pred_acc,
                                    const float* __restrict__ wsum_acc,
                                    float* __restrict__ out) {
    int idx = blockIdx.x * blockDim.x + threadIdx.x;  // 0 .. 2048*16-1
    int r = idx >> 4;
    out[idx] = pred_acc[idx] / (wsum_acc[r] + 1e-10f);
}

// ---------------------------------------------------------------------------
extern "C" void kernel_launch(void* const* d_in, const int* in_sizes, int n_in,
                              void* d_out, int out_size, void* d_ws,
                              size_t ws_size, hipStream_t stream) {
    (void)in_sizes; (void)n_in; (void)out_size; (void)ws_size;
    const float* x  = (const float*)d_in[0];   // [2048, 64]
    const float* tx = (const float*)d_in[1];   // [50000, 64]
    const float* ty = (const float*)d_in[2];   // [50000, 16]
    float* out = (float*)d_out;                // [2048, 16]

    char* ws = (char*)d_ws;
    float* pred_acc = (float*)ws;                                  // 128 KB
    float* wsum_acc = (float*)(ws + (size_t)B_ROWS * COLS * 4);    // 8 KB
    float* tsq = (float*)(ws + (size_t)B_ROWS * COLS * 4 + B_ROWS * 4); // 200 KB

    // Zero the accumulators (tsq fully overwritten, no need).
    hipMemsetAsync(ws, 0, (size_t)B_ROWS * COLS * 4 + B_ROWS * 4, stream);

    pnn_tsq_kernel<<<(N_TRAIN + 255) / 256, 256, 0, stream>>>(tx, tsq, N_TRAIN);

    dim3 grid(B_ROWS / 32, SLICES);   // 64 tile-pairs x 125 train slices
    pnn_main_kernel<<<grid, 32, 0, stream>>>(x, tx, ty, tsq, pred_acc, wsum_acc);

    pnn_finalize_kernel<<<(B_ROWS * COLS) / 256, 256, 0, stream>>>(
        pred_acc, wsum_acc, out);
}